// SwitchRouter_61229053772308
// MI455X (gfx1250) — compile-verified
//
#include <hip/hip_runtime.h>
#include <cmath>

// ---------------------------------------------------------------------------
// SwitchRouter for MI455X (gfx1250, wave32)
//   logits = hidden[32768,4096] @ W[64,4096]^T  via V_WMMA_F32_16X16X4_F32
//   then softmax/argmax/load-variance/entropy.
// ---------------------------------------------------------------------------

typedef __attribute__((ext_vector_type(2))) float v2f;
typedef __attribute__((ext_vector_type(8))) float v8f;

#define T_TOK 32768
#define HDIM  4096
#define NEXP  64

// ---------------------------------------------------------------------------
// Kernel 1: router GEMM.  Each wave owns a 32(M) x 64(N=experts) tile.
// A fragment (16x4 f32, per ISA layout): lane L<16 holds A[M=L][K=k..k+1] in
// v0/v1, lane L+16 holds A[M=L][K=k+2..k+3]  -> one aligned float2 per lane.
// B fragment (4x16) mirrors this with N across lanes: W[n][k + 2*half ..+1].
// C/D: VGPR r, lanes 0-15 -> M=r, lanes 16-31 -> M=r+8, N = lane&15.
// ---------------------------------------------------------------------------
__global__ __launch_bounds__(256)
void router_gemm_wmma(const float* __restrict__ A,   // [T_TOK, HDIM]
                      const float* __restrict__ W,   // [NEXP, HDIM]
                      float* __restrict__ logits) {  // [T_TOK, NEXP]
    const int lane = threadIdx.x & 31;
    const int wave = threadIdx.x >> 5;
    const int half = lane >> 4;       // 0 or 1: selects K base 0 / 2
    const int l16  = lane & 15;
    const int rowBase = blockIdx.x * 256 + wave * 32;

    const float* aptr0 = A + (size_t)(rowBase + l16) * HDIM + 2 * half;
    const float* aptr1 = aptr0 + (size_t)16 * HDIM;
    const float* bptr0 = W + (size_t)(0 * 16 + l16) * HDIM + 2 * half;
    const float* bptr1 = W + (size_t)(1 * 16 + l16) * HDIM + 2 * half;
    const float* bptr2 = W + (size_t)(2 * 16 + l16) * HDIM + 2 * half;
    const float* bptr3 = W + (size_t)(3 * 16 + l16) * HDIM + 2 * half;

    v8f acc[2][4];
#pragma unroll
    for (int mt = 0; mt < 2; ++mt)
#pragma unroll
        for (int nt = 0; nt < 4; ++nt)
            acc[mt][nt] = (v8f){0.f, 0.f, 0.f, 0.f, 0.f, 0.f, 0.f, 0.f};

    for (int k = 0; k < HDIM; k += 4) {
        // Stream-ahead prefetch of the A rows (uniform branch, scalar).
        if ((k & 127) == 0 && (k + 512) < HDIM) {
            __builtin_prefetch(aptr0 + k + 512, 0, 3);
            __builtin_prefetch(aptr1 + k + 512, 0, 3);
        }
        v2f a0 = *(const v2f*)(aptr0 + k);
        v2f a1 = *(const v2f*)(aptr1 + k);
        v2f b0 = *(const v2f*)(bptr0 + k);
        v2f b1 = *(const v2f*)(bptr1 + k);
        v2f b2 = *(const v2f*)(bptr2 + k);
        v2f b3 = *(const v2f*)(bptr3 + k);

        acc[0][0] = __builtin_amdgcn_wmma_f32_16x16x4_f32(false, a0, false, b0, (short)0, acc[0][0], false, false);
        acc[0][1] = __builtin_amdgcn_wmma_f32_16x16x4_f32(false, a0, false, b1, (short)0, acc[0][1], false, false);
        acc[0][2] = __builtin_amdgcn_wmma_f32_16x16x4_f32(false, a0, false, b2, (short)0, acc[0][2], false, false);
        acc[0][3] = __builtin_amdgcn_wmma_f32_16x16x4_f32(false, a0, false, b3, (short)0, acc[0][3], false, false);
        acc[1][0] = __builtin_amdgcn_wmma_f32_16x16x4_f32(false, a1, false, b0, (short)0, acc[1][0], false, false);
        acc[1][1] = __builtin_amdgcn_wmma_f32_16x16x4_f32(false, a1, false, b1, (short)0, acc[1][1], false, false);
        acc[1][2] = __builtin_amdgcn_wmma_f32_16x16x4_f32(false, a1, false, b2, (short)0, acc[1][2], false, false);
        acc[1][3] = __builtin_amdgcn_wmma_f32_16x16x4_f32(false, a1, false, b3, (short)0, acc[1][3], false, false);
    }

    // Writeback: C layout -> lanes 0-15: M=r, lanes 16-31: M=r+8; N = l16.
#pragma unroll
    for (int mt = 0; mt < 2; ++mt)
#pragma unroll
        for (int r = 0; r < 8; ++r) {
            const size_t row = (size_t)(rowBase + mt * 16 + r + 8 * half);
#pragma unroll
            for (int nt = 0; nt < 4; ++nt)
                logits[row * NEXP + nt * 16 + l16] = acc[mt][nt][r];
        }
}

// ---------------------------------------------------------------------------
// Kernel 2: per-token softmax / argmax / entropy + deterministic block
// partial sums for expert load & entropy (fixed-order shuffle trees, wave32).
// ---------------------------------------------------------------------------
__global__ __launch_bounds__(256)
void router_softmax(const float* __restrict__ logits,
                    float* __restrict__ sel,        // [T_TOK] (as float)
                    float* __restrict__ wts,        // [T_TOK]
                    float* __restrict__ part_load,  // [nblocks, NEXP]
                    float* __restrict__ part_ent) { // [nblocks]
    __shared__ float s_red[NEXP][8];
    __shared__ float s_ent[8];

    const int t    = blockIdx.x * 256 + threadIdx.x;
    const int lane = threadIdx.x & 31;
    const int wave = threadIdx.x >> 5;

    float p[NEXP];
    const float* lp = logits + (size_t)t * NEXP;
#pragma unroll
    for (int e = 0; e < NEXP; ++e) p[e] = lp[e];

    float m = p[0];
#pragma unroll
    for (int e = 1; e < NEXP; ++e) m = fmaxf(m, p[e]);

    float sum = 0.f;
#pragma unroll
    for (int e = 0; e < NEXP; ++e) {
        float v = expf(p[e] - m);
        p[e] = v;
        sum += v;
    }
    const float inv = 1.f / sum;
#pragma unroll
    for (int e = 0; e < NEXP; ++e) p[e] *= inv;

    // First-index argmax over probs (matches jnp.argmax tie rule).
    int amax = 0;
    float pmax = p[0];
#pragma unroll
    for (int e = 1; e < NEXP; ++e)
        if (p[e] > pmax) { pmax = p[e]; amax = e; }

    float ent = 0.f;
#pragma unroll
    for (int e = 0; e < NEXP; ++e) ent -= p[e] * logf(p[e] + 1e-8f);

    sel[t] = (float)amax;
    wts[t] = pmax;

    // Deterministic reductions: wave32 xor-shuffle tree, then 8-wave combine.
#pragma unroll
    for (int e = 0; e < NEXP; ++e) {
        float v = p[e];
        v += __shfl_xor(v, 16);
        v += __shfl_xor(v, 8);
        v += __shfl_xor(v, 4);
        v += __shfl_xor(v, 2);
        v += __shfl_xor(v, 1);
        if (lane == 0) s_red[e][wave] = v;
    }
    {
        float v = ent;
        v += __shfl_xor(v, 16);
        v += __shfl_xor(v, 8);
        v += __shfl_xor(v, 4);
        v += __shfl_xor(v, 2);
        v += __shfl_xor(v, 1);
        if (lane == 0) s_ent[wave] = v;
    }
    __syncthreads();

    if (threadIdx.x < NEXP) {
        float tot = 0.f;
#pragma unroll
        for (int w = 0; w < 8; ++w) tot += s_red[threadIdx.x][w];
        part_load[(size_t)blockIdx.x * NEXP + threadIdx.x] = tot;
    }
    if (threadIdx.x == 0) {
        float tot = 0.f;
#pragma unroll
        for (int w = 0; w < 8; ++w) tot += s_ent[w];
        part_ent[blockIdx.x] = tot;
    }
}

// ---------------------------------------------------------------------------
// Kernel 3: fixed-order final reduction -> load variance (ddof=1) + entropy.
// ---------------------------------------------------------------------------
__global__ __launch_bounds__(64)
void router_finalize(const float* __restrict__ part_load,
                     const float* __restrict__ part_ent,
                     int nblocks,
                     float* __restrict__ var_out,
                     float* __restrict__ ent_out) {
    __shared__ float s_load[NEXP];
    const int e = threadIdx.x;
    float tot = 0.f;
    for (int b = 0; b < nblocks; ++b) tot += part_load[(size_t)b * NEXP + e];
    s_load[e] = tot / (float)T_TOK;   // expert_load[e] = mean prob
    __syncthreads();

    if (e == 0) {
        float mean = 0.f;
        for (int i = 0; i < NEXP; ++i) mean += s_load[i];
        mean /= (float)NEXP;
        float var = 0.f;
        for (int i = 0; i < NEXP; ++i) {
            const float d = s_load[i] - mean;
            var += d * d;
        }
        var /= (float)(NEXP - 1);     // unbiased (torch .var default)
        *var_out = var;

        float entsum = 0.f;
        for (int b = 0; b < nblocks; ++b) entsum += part_ent[b];
        *ent_out = entsum / (float)T_TOK;
    }
}

// ---------------------------------------------------------------------------
extern "C" void kernel_launch(void* const* d_in, const int* in_sizes, int n_in,
                              void* d_out, int out_size, void* d_ws, size_t ws_size,
                              hipStream_t stream) {
    (void)in_sizes; (void)n_in; (void)out_size; (void)ws_size;

    const float* hidden = (const float*)d_in[0];   // [32768, 4096] f32
    const float* weight = (const float*)d_in[1];   // [64, 4096]    f32

    float* out     = (float*)d_out;
    float* logits  = out;                                   // T*64
    float* sel     = out + (size_t)T_TOK * NEXP;            // T
    float* wts     = sel + T_TOK;                           // T
    float* var_out = wts + T_TOK;                           // 1
    float* ent_out = var_out + 1;                           // 1

    const int nblk = T_TOK / 256;                           // 128
    float* ws        = (float*)d_ws;
    float* part_load = ws;                                  // nblk*64 floats
    float* part_ent  = ws + (size_t)nblk * NEXP;            // nblk floats

    router_gemm_wmma<<<T_TOK / 256, 256, 0, stream>>>(hidden, weight, logits);
    router_softmax<<<nblk, 256, 0, stream>>>(logits, sel, wts, part_load, part_ent);
    router_finalize<<<1, 64, 0, stream>>>(part_load, part_ent, nblk, var_out, ent_out);
}